// QuantumNet_12369505813245
// MI455X (gfx1250) — compile-verified
//
#include <hip/hip_runtime.h>
#include <math.h>

#define N_Q   10
#define DIM   1024
#define NPTS  128
#define NPAIR (NPTS * NPTS)

typedef __attribute__((ext_vector_type(2))) float v2f;
typedef __attribute__((ext_vector_type(8))) float v8f;

__device__ __forceinline__ int bitq(int d, int q) { return (d >> (9 - q)) & 1; }
// bit-swap hi/lo 5 bits so diag-table reads coalesce as tab[t*32+lane]
__device__ __forceinline__ int permd(int d) { return ((d & 31) << 5) | (d >> 5); }

// ---------------------------------------------------------------------------
// Kernel 1: RZ phase tables via WMMA f32 16x16x4.
// angle[d,i,b] = sum_q (bit(d,q)-0.5) * data[i, b*10+q]
// M = amplitude tile (16), N = data-point tile (16), K = 10 qubits padded to 12.
// ---------------------------------------------------------------------------
__global__ __launch_bounds__(256)
void rz_phase_kernel(const float* __restrict__ data,
                     float* __restrict__ rz_cos, float* __restrict__ rz_sin) {
    const int wid  = blockIdx.x * (blockDim.x >> 5) + (threadIdx.x >> 5);
    const int lane = threadIdx.x & 31;
    const int b  = wid / 512;          // block index 0..9
    const int rem = wid - b * 512;
    const int td = rem >> 3;           // d-tile 0..63
    const int ti = rem & 7;            // i-tile 0..7
    if (b >= N_Q) return;

    const int half = lane >> 4;        // 0: lanes 0-15, 1: lanes 16-31
    const int lm   = lane & 15;

    v8f acc = {0.f, 0.f, 0.f, 0.f, 0.f, 0.f, 0.f, 0.f};
    const int dA = td * 16 + lm;       // A-fragment row (amplitude)
    const int iB = ti * 16 + lm;       // B-fragment col (data point)

    #pragma unroll
    for (int kk = 0; kk < 3; ++kk) {
        const int k0 = kk * 4 + half * 2;   // K indices this lane supplies
        v2f a, bb;
        a.x  = (k0     < N_Q) ? ((float)bitq(dA, k0)     - 0.5f) : 0.0f;
        a.y  = (k0 + 1 < N_Q) ? ((float)bitq(dA, k0 + 1) - 0.5f) : 0.0f;
        bb.x = (k0     < N_Q) ? data[iB * 100 + b * 10 + k0]     : 0.0f;
        bb.y = (k0 + 1 < N_Q) ? data[iB * 100 + b * 10 + k0 + 1] : 0.0f;
        acc = __builtin_amdgcn_wmma_f32_16x16x4_f32(false, a, false, bb,
                                                    (short)0, acc, false, false);
    }

    // C/D layout: vgpr r, lanes0-15 -> M=r, lanes16-31 -> M=r+8; N = lane%16
    #pragma unroll
    for (int r = 0; r < 8; ++r) {
        const int d = td * 16 + r + half * 8;
        float sv, cv;
        __sincosf(acc[r], &sv, &cv);
        const int idx = (iB * N_Q + b) * DIM + permd(d);
        rz_cos[idx] = cv;
        rz_sin[idx] = sv;
    }
}

// ---------------------------------------------------------------------------
// Kernel 2: CRZ-ring phase tables (depend only on params) + RY cos/sin.
// ---------------------------------------------------------------------------
__global__ __launch_bounds__(256)
void crz_ry_kernel(const float* __restrict__ params,
                   float* __restrict__ crz_cos, float* __restrict__ crz_sin,
                   float* __restrict__ ryc, float* __restrict__ rys) {
    const int t = blockIdx.x * blockDim.x + threadIdx.x;
    if (t < N_Q * DIM) {
        const int b = t >> 10, d = t & (DIM - 1);
        float ang = 0.f;
        #pragma unroll
        for (int q = 0; q < N_Q; ++q) {
            const float ring = (float)bitq(d, q) *
                               ((float)bitq(d, (q + 1) % N_Q) - 0.5f);
            ang = fmaf(ring, params[b * 20 + 10 + q], ang);
        }
        float sv, cv;
        __sincosf(ang, &sv, &cv);
        const int idx = b * DIM + permd(d);
        crz_cos[idx] = cv;
        crz_sin[idx] = sv;
    }
    if (t < N_Q * N_Q) {
        const int b = t / N_Q, q = t - b * N_Q;
        const float th = params[b * 20 + q] * 0.5f;
        ryc[t] = __cosf(th);
        rys[t] = __sinf(th);
    }
}

// ---------------------------------------------------------------------------
// Kernel 3: state-vector simulation. One pair per wave32.
// Lane holds 32 complex amps: d = (lane<<5) | t.
// ---------------------------------------------------------------------------
__device__ __forceinline__ void h_layer(float (&re)[32], float (&im)[32], int lane) {
    // cross-lane qubits (d bits 9..5 == lane bits 4..0)
    #pragma unroll
    for (int lb = 0; lb < 5; ++lb) {
        const int mask = 1 << lb;
        const float sgn = ((lane >> lb) & 1) ? -1.0f : 1.0f;
        #pragma unroll
        for (int t = 0; t < 32; ++t) {
            const float pr = __shfl_xor(re[t], mask, 32);
            const float pi = __shfl_xor(im[t], mask, 32);
            re[t] = fmaf(sgn, re[t], pr);
            im[t] = fmaf(sgn, im[t], pi);
        }
    }
    // in-lane qubits (d bits 4..0)
    #pragma unroll
    for (int bp = 0; bp < 5; ++bp) {
        const int m = 1 << bp;
        #pragma unroll
        for (int t = 0; t < 32; ++t) {
            if (!(t & m)) {
                const float a = re[t], bb = re[t | m];
                re[t] = a + bb;  re[t | m] = a - bb;
                const float ai = im[t], bi = im[t | m];
                im[t] = ai + bi; im[t | m] = ai - bi;
            }
        }
    }
    #pragma unroll
    for (int t = 0; t < 32; ++t) { re[t] *= 0.03125f; im[t] *= 0.03125f; }
}

__device__ __forceinline__ void ry_layer(float (&re)[32], float (&im)[32], int lane,
                                         const float* __restrict__ ryc,
                                         const float* __restrict__ rys,
                                         int b, float ssign) {
    #pragma unroll
    for (int q = 0; q < N_Q; ++q) {
        const int bp = 9 - q;                 // bit position of qubit q
        const float c = ryc[b * N_Q + q];
        const float s = ssign * rys[b * N_Q + q];
        if (bp >= 5) {
            const int lb = bp - 5;
            const int mask = 1 << lb;
            const float ss = ((lane >> lb) & 1) ? s : -s;
            #pragma unroll
            for (int t = 0; t < 32; ++t) {
                const float pr = __shfl_xor(re[t], mask, 32);
                const float pi = __shfl_xor(im[t], mask, 32);
                re[t] = fmaf(c, re[t], ss * pr);
                im[t] = fmaf(c, im[t], ss * pi);
            }
        } else {
            const int m = 1 << bp;
            #pragma unroll
            for (int t = 0; t < 32; ++t) {
                if (!(t & m)) {
                    const float a0 = re[t], a1 = re[t | m];
                    re[t]     = fmaf(c, a0, -s * a1);
                    re[t | m] = fmaf(c, a1,  s * a0);
                    const float b0 = im[t], b1 = im[t | m];
                    im[t]     = fmaf(c, b0, -s * b1);
                    im[t | m] = fmaf(c, b1,  s * b0);
                }
            }
        }
    }
}

__device__ __forceinline__ void diag_layer(float (&re)[32], float (&im)[32], int lane,
                                           const float* __restrict__ ctab,
                                           const float* __restrict__ stab,
                                           float ssign) {
    #pragma unroll
    for (int t = 0; t < 32; ++t) {
        const float c = ctab[t * 32 + lane];          // coalesced (perm layout)
        const float s = ssign * stab[t * 32 + lane];
        const float r = re[t], ii = im[t];
        re[t] = r * c - ii * s;
        im[t] = fmaf(r, s, ii * c);
    }
}

__global__ __launch_bounds__(256)
void sim_kernel(const float* __restrict__ rz_cos, const float* __restrict__ rz_sin,
                const float* __restrict__ crz_cos, const float* __restrict__ crz_sin,
                const float* __restrict__ ryc, const float* __restrict__ rys,
                float* __restrict__ Kmat) {
    const int lane = threadIdx.x & 31;
    const int pair = blockIdx.x * (blockDim.x >> 5) + (threadIdx.x >> 5);
    const int i = pair >> 7;          // x1 point (pair-major)
    const int j = pair & 127;         // x2 point

    float re[32], im[32];
    #pragma unroll
    for (int t = 0; t < 32; ++t) { re[t] = 0.f; im[t] = 0.f; }
    re[0] = (lane == 0) ? 1.0f : 0.0f;

    #pragma unroll 1
    for (int b = 0; b < N_Q; ++b) {   // forward blocks
        h_layer(re, im, lane);
        diag_layer(re, im, lane, rz_cos + (i * N_Q + b) * DIM,
                                 rz_sin + (i * N_Q + b) * DIM, 1.0f);
        ry_layer(re, im, lane, ryc, rys, b, 1.0f);
        diag_layer(re, im, lane, crz_cos + b * DIM, crz_sin + b * DIM, 1.0f);
    }
    #pragma unroll 1
    for (int k = 0; k < N_Q; ++k) {   // adjoint blocks (reversed params)
        const int b = 9 - k;          // also the x2 block: start=(9-k)*10
        diag_layer(re, im, lane, crz_cos + b * DIM, crz_sin + b * DIM, -1.0f);
        ry_layer(re, im, lane, ryc, rys, b, -1.0f);
        diag_layer(re, im, lane, rz_cos + (j * N_Q + b) * DIM,
                                 rz_sin + (j * N_Q + b) * DIM, -1.0f);
        h_layer(re, im, lane);
    }
    if (lane == 0) Kmat[pair] = re[0] * re[0] + im[0] * im[0];
}

// ---------------------------------------------------------------------------
// Kernel 4: kernel-target-alignment reduction -> scalar.
// ---------------------------------------------------------------------------
__global__ __launch_bounds__(256)
void reduce_kernel(const float* __restrict__ Kmat, const float* __restrict__ labels,
                   float* __restrict__ out) {
    __shared__ float sp[256], sk[256];
    float pol = 0.f, k2 = 0.f;
    for (int idx = threadIdx.x; idx < NPAIR; idx += 256) {
        const float Kv = Kmat[idx];
        const float ll = labels[idx >> 7] * labels[idx & 127];
        pol = fmaf(ll, Kv, pol);
        k2  = fmaf(Kv, Kv, k2);
    }
    sp[threadIdx.x] = pol; sk[threadIdx.x] = k2;
    __syncthreads();
    for (int off = 128; off > 0; off >>= 1) {
        if ((int)threadIdx.x < off) {
            sp[threadIdx.x] += sp[threadIdx.x + off];
            sk[threadIdx.x] += sk[threadIdx.x + off];
        }
        __syncthreads();
    }
    if (threadIdx.x == 0) {
        float sl2 = 0.f;
        for (int t = 0; t < NPTS; ++t) { const float l = labels[t]; sl2 = fmaf(l, l, sl2); }
        out[0] = sp[0] / sqrtf(sk[0] * sl2 * sl2);   // sum(ll^2) == (sum l^2)^2
    }
}

// ---------------------------------------------------------------------------
extern "C" void kernel_launch(void* const* d_in, const int* in_sizes, int n_in,
                              void* d_out, int out_size, void* d_ws, size_t ws_size,
                              hipStream_t stream) {
    const float* data   = (const float*)d_in[0];   // [128, 100]
    const float* labels = (const float*)d_in[1];   // [128]
    const float* params = (const float*)d_in[2];   // [10, 2, 10]

    float* ws      = (float*)d_ws;
    float* rz_cos  = ws;                               // 128*10*1024
    float* rz_sin  = rz_cos  + NPTS * N_Q * DIM;       // 128*10*1024
    float* crz_cos = rz_sin  + NPTS * N_Q * DIM;       // 10*1024
    float* crz_sin = crz_cos + N_Q * DIM;              // 10*1024
    float* ryc     = crz_sin + N_Q * DIM;              // 128 (padded)
    float* rys     = ryc + 128;                        // 128 (padded)
    float* Kmat    = rys + 128;                        // 16384

    // 10 blocks * 64 d-tiles * 8 i-tiles = 5120 waves, 8 waves/block
    rz_phase_kernel<<<640, 256, 0, stream>>>(data, rz_cos, rz_sin);
    crz_ry_kernel<<<40, 256, 0, stream>>>(params, crz_cos, crz_sin, ryc, rys);
    sim_kernel<<<NPAIR / 8, 256, 0, stream>>>(rz_cos, rz_sin, crz_cos, crz_sin,
                                              ryc, rys, Kmat);
    reduce_kernel<<<1, 256, 0, stream>>>(Kmat, labels, (float*)d_out);
}